// GNN_412316860424
// MI455X (gfx1250) — compile-verified
//
#include <hip/hip_runtime.h>

// ---------------------------------------------------------------------------
// GNN message passing:  h = scatter_add( relu(x @ W + b)[src] -> dst )  x2,
// then out = h @ Wf + bf.
// Optimization: gather commutes with row-wise linear+relu, so the per-edge
// MLP collapses to a per-node GEMM (16x fewer FLOPs) + pure scatter-add.
// GEMMs use V_WMMA_F32_16X16X4_F32 (exact fp32, matches reference).
// ---------------------------------------------------------------------------

typedef float v2f __attribute__((ext_vector_type(2)));
typedef float v8f __attribute__((ext_vector_type(8)));

#define HID 128

// T[N,128] = relu(X[N,128] @ W[128,128] + b[128])
// grid.x = N/16 (N multiple of 16), block = 256 (8 waves).
// wave w computes the 16x16 tile at (m_base = blockIdx.x*16, n_base = w*16).
__global__ __launch_bounds__(256)
void gnn_linear_relu_wmma(const float* __restrict__ X,
                          const float* __restrict__ W,
                          const float* __restrict__ bias,
                          float* __restrict__ T) {
    const int lane   = threadIdx.x & 31;
    const int wave   = threadIdx.x >> 5;
    const int m_base = blockIdx.x << 4;
    const int n_base = wave << 4;        // 8 waves cover 128 columns
    const int mr     = lane & 15;        // row-in-tile / col-in-tile
    const int kh     = lane >> 4;        // K half-select (0 or 1)

    const float* __restrict__ xrow = X + (size_t)(m_base + mr) * HID;
    const int col = n_base + mr;

    v8f acc = {0.f, 0.f, 0.f, 0.f, 0.f, 0.f, 0.f, 0.f};

    // A 16x4 fp32 fragment: lanes 0-15 hold K={k0,k0+1}, lanes 16-31 K={k0+2,k0+3}
    // B 4x16 fp32 fragment: same K striping, N = lane%16.
    #pragma unroll 4
    for (int k0 = 0; k0 < HID; k0 += 4) {
        const int k = k0 + (kh << 1);
        v2f a;
        a.x = xrow[k];
        a.y = xrow[k + 1];
        v2f b;
        b.x = W[(size_t)k * HID + col];
        b.y = W[(size_t)(k + 1) * HID + col];
        acc = __builtin_amdgcn_wmma_f32_16x16x4_f32(
            /*neg_a=*/false, a, /*neg_b=*/false, b,
            /*c_mod=*/(short)0, acc, /*reuse_a=*/false, /*reuse_b=*/false);
    }

    // C/D layout: VGPR r -> row (r + 8*kh), col = lane%16
    const float bv = bias[col];
    #pragma unroll
    for (int r = 0; r < 8; ++r) {
        const int row = m_base + r + (kh << 3);
        const float v = acc[r] + bv;
        T[(size_t)row * HID + col] = fmaxf(v, 0.0f);
    }
}

__global__ void gnn_zero_f4(float4* __restrict__ p, int n4) {
    const int i = blockIdx.x * blockDim.x + threadIdx.x;
    if (i < n4) p[i] = make_float4(0.f, 0.f, 0.f, 0.f);
}

// 32 lanes per edge; each lane moves 4 floats: gather T[src], atomic-add into H[dst].
// Both T and H (25.6 MB each) are L2-resident on MI455X (192 MB L2).
__global__ __launch_bounds__(256)
void gnn_scatter_add(const float* __restrict__ T,
                     const int* __restrict__ src,
                     const int* __restrict__ dst,
                     float* __restrict__ H, int E) {
    const int idx = blockIdx.x * blockDim.x + threadIdx.x;
    const int e = idx >> 5;
    if (e >= E) return;
    const int f = (idx & 31) << 2;
    const int s = src[e];
    const int d = dst[e];
    const float4 v = *(const float4*)(T + (size_t)s * HID + f);
    float* hp = H + (size_t)d * HID + f;
    atomicAdd(hp + 0, v.x);
    atomicAdd(hp + 1, v.y);
    atomicAdd(hp + 2, v.z);
    atomicAdd(hp + 3, v.w);
}

// out[n] = dot(H[n,:], Wf[:,0]) + bf   — one wave per node, wave32 reduction.
__global__ __launch_bounds__(256)
void gnn_final_gemv(const float* __restrict__ H,
                    const float* __restrict__ Wf,
                    const float* __restrict__ bf,
                    float* __restrict__ out, int N) {
    const int wid  = (blockIdx.x * blockDim.x + threadIdx.x) >> 5;
    const int lane = threadIdx.x & 31;
    if (wid >= N) return;
    const float4 v = *(const float4*)(H + (size_t)wid * HID + (lane << 2));
    const float4 w = *(const float4*)(Wf + (lane << 2));
    float s = v.x * w.x + v.y * w.y + v.z * w.z + v.w * w.w;
    #pragma unroll
    for (int off = 16; off > 0; off >>= 1)
        s += __shfl_xor(s, off, 32);
    if (lane == 0) out[wid] = s + bf[0];
}

extern "C" void kernel_launch(void* const* d_in, const int* in_sizes, int n_in,
                              void* d_out, int out_size, void* d_ws, size_t ws_size,
                              hipStream_t stream) {
    const float* x   = (const float*)d_in[0];
    const int*   ei  = (const int*)  d_in[1];   // edge_index [2, E]
    const float* W1  = (const float*)d_in[2];
    const float* b1  = (const float*)d_in[3];
    const float* W2  = (const float*)d_in[4];
    const float* b2  = (const float*)d_in[5];
    const float* Wf  = (const float*)d_in[6];
    const float* bf  = (const float*)d_in[7];
    float* out = (float*)d_out;

    const int N = in_sizes[0] / HID;   // 50000
    const int E = in_sizes[1] / 2;     // 800000
    const int* src = ei;
    const int* dst = ei + E;

    float* bufT = (float*)d_ws;                 // [N, HID] transformed messages
    float* bufH = bufT + (size_t)N * HID;       // [N, HID] aggregated features

    const dim3 blk(256);
    const int n4        = (N * HID) / 4;
    const int zeroGrid  = (n4 + 255) / 256;
    const int gemmGrid  = N / 16;               // N = 50000 = 3125 * 16
    const int scatGrid  = (int)(((size_t)E * 32 + 255) / 256);
    const int gemvGrid  = (N * 32 + 255) / 256;

    // Layer 1: T = relu(x @ W1 + b1); H = scatter_add(T[src] -> dst)
    gnn_linear_relu_wmma<<<gemmGrid, blk, 0, stream>>>(x, W1, b1, bufT);
    gnn_zero_f4<<<zeroGrid, blk, 0, stream>>>((float4*)bufH, n4);
    gnn_scatter_add<<<scatGrid, blk, 0, stream>>>(bufT, src, dst, bufH, E);

    // Layer 2: T = relu(H @ W2 + b2); H = scatter_add(T[src] -> dst)
    gnn_linear_relu_wmma<<<gemmGrid, blk, 0, stream>>>(bufH, W2, b2, bufT);
    gnn_zero_f4<<<zeroGrid, blk, 0, stream>>>((float4*)bufH, n4);
    gnn_scatter_add<<<scatGrid, blk, 0, stream>>>(bufT, src, dst, bufH, E);

    // Readout: out = H @ Wf + bf
    gnn_final_gemv<<<gemvGrid, blk, 0, stream>>>(bufH, Wf, bf, out, N);
}